// VSGCLayerPre_11914239279381
// MI455X (gfx1250) — compile-verified
//
#include <hip/hip_runtime.h>
#include <hip/hip_bf16.h>

typedef __attribute__((ext_vector_type(2))) float v2f;
typedef __attribute__((ext_vector_type(8))) float v8f;

#define IN_DIM 128
#define OUT_DIM 64
#define ALPHA 0.5f
#define LAMBD 1.0f

// ---------------- utility: zero a float buffer ----------------
__global__ void vsgc_zero_kernel(float* __restrict__ p, long long n) {
    long long i = (long long)blockIdx.x * blockDim.x + threadIdx.x;
    if (i < n) p[i] = 0.0f;
}

// ---------------- degree count ----------------
__global__ void vsgc_deg_kernel(const int* __restrict__ dst, float* __restrict__ degs, int E) {
    int e = blockIdx.x * blockDim.x + threadIdx.x;
    if (e < E) atomicAdd(&degs[dst[e]], 1.0f);
}

// ---------------- per-node norms ----------------
__global__ void vsgc_norm_kernel(const float* __restrict__ degs,
                                 float* __restrict__ nl1,   // 1/(lambda*deg+1)
                                 float* __restrict__ n05,   // sqrt(deg+1)
                                 float* __restrict__ n_05,  // rsqrt(deg+1)
                                 int N) {
    int i = blockIdx.x * blockDim.x + threadIdx.x;
    if (i < N) {
        float d = degs[i];
        nl1[i]  = 1.0f / (LAMBD * d + 1.0f);
        float dp1 = d + 1.0f;
        n05[i]  = sqrtf(dp1);
        n_05[i] = rsqrtf(dp1);
    }
}

// ---------------- GEMM: h = features @ W^T + b  via V_WMMA_F32_16X16X4_F32 ----
// Block = 128 threads = 4 wave32 waves. Each wave computes one 16x16 tile of h:
// rows [16*blockIdx.x, +16), cols [16*wave, +16). K=128 -> 32 chained WMMAs.
// Also emits h_init = h * nl1 (fused).
__global__ __launch_bounds__(128)
void vsgc_gemm_wmma_kernel(const float* __restrict__ feat,
                           const float* __restrict__ W,
                           const float* __restrict__ bias,
                           const float* __restrict__ nl1,
                           float* __restrict__ h,
                           float* __restrict__ h_init) {
    const int wave = threadIdx.x >> 5;     // 0..3 -> column tile
    const int lane = threadIdx.x & 31;
    const int half = lane >> 4;            // 0: K even pair / M 0..7 ; 1: K odd pair / M 8..15
    const int lrow = lane & 15;
    const int row0 = blockIdx.x * 16;
    const int n0   = wave * 16;

    // A: lane<16 -> (M=lrow, K={4k,4k+1}); lane>=16 -> (M=lrow, K={4k+2,4k+3})
    const float* arow = feat + (size_t)(row0 + lrow) * IN_DIM + 2 * half;
    // B[k][n] = W[n][k]; same striping over K as A
    const float* brow = W    + (size_t)(n0   + lrow) * IN_DIM + 2 * half;

    v8f c = {};
#pragma unroll
    for (int kq = 0; kq < IN_DIM / 4; ++kq) {
        v2f a = *(const v2f*)(arow + kq * 4);
        v2f b = *(const v2f*)(brow + kq * 4);
        // 8 args: (neg_a, A, neg_b, B, c_mod, C, reuse_a, reuse_b)
        c = __builtin_amdgcn_wmma_f32_16x16x4_f32(false, a, false, b, (short)0, c,
                                                  false, false);
    }

    const float bb = bias[n0 + lrow];
#pragma unroll
    for (int v = 0; v < 8; ++v) {
        int m = row0 + v + 8 * half;       // C/D layout: VGPR v holds M = v + 8*half
        float val = c[v] + bb;
        size_t idx = (size_t)m * OUT_DIM + n0 + lrow;
        h[idx]      = val;
        h_init[idx] = val * nl1[m];
    }
}

// ---------------- edge scatter: agg[dst] += h[src] * n_05[src] ----------------
// 64 consecutive threads handle one edge (one thread per feature dim):
// contiguous 256B gathers and contiguous atomic bursts into L2.
__global__ void vsgc_scatter_kernel(const int* __restrict__ src,
                                    const int* __restrict__ dst,
                                    const float* __restrict__ h,
                                    const float* __restrict__ n_05,
                                    float* __restrict__ agg,
                                    long long total) {
    long long i = (long long)blockIdx.x * blockDim.x + threadIdx.x;
    if (i >= total) return;
    int e = (int)(i >> 6);
    int d = (int)(i & 63);
    int s = src[e];
    int t = dst[e];
    float v = h[(size_t)s * OUT_DIM + d] * n_05[s];
    atomicAdd(&agg[(size_t)t * OUT_DIM + d], v);
}

// ---------------- blend: hn = 0.5*(agg*nl1*n05) + 0.5*h + 0.5*h_init ----------
__global__ void vsgc_update_kernel(const float* __restrict__ agg,
                                   const float* __restrict__ h,
                                   const float* __restrict__ h_init,
                                   const float* __restrict__ nl1,
                                   const float* __restrict__ n05,
                                   float* __restrict__ out,
                                   long long total) {
    long long i = (long long)blockIdx.x * blockDim.x + threadIdx.x;
    if (i >= total) return;
    int m = (int)(i >> 6);
    float h2 = agg[i] * nl1[m] * n05[m];
    out[i] = (ALPHA * LAMBD) * h2 + (1.0f - ALPHA) * h[i] + ALPHA * h_init[i];
}

extern "C" void kernel_launch(void* const* d_in, const int* in_sizes, int n_in,
                              void* d_out, int out_size, void* d_ws, size_t ws_size,
                              hipStream_t stream) {
    const float* feat = (const float*)d_in[0];   // N x 128
    const float* W    = (const float*)d_in[1];   // 64 x 128
    const float* bias = (const float*)d_in[2];   // 64
    const int*   src  = (const int*)d_in[3];     // E
    const int*   dst  = (const int*)d_in[4];     // E

    const int N = in_sizes[0] / IN_DIM;
    const int E = in_sizes[3];
    const long long NF = (long long)N * OUT_DIM;

    float* ws     = (float*)d_ws;
    float* h      = ws;                 // N*64
    float* h_init = h      + NF;        // N*64
    float* agg    = h_init + NF;        // N*64
    float* degs   = agg    + NF;        // N
    float* nl1    = degs   + N;         // N
    float* n05    = nl1    + N;         // N
    float* n_05   = n05    + N;         // N

    const int T = 256;

    // 1) degrees
    vsgc_zero_kernel<<<(int)((N + T - 1) / T), T, 0, stream>>>(degs, N);
    vsgc_deg_kernel<<<(E + T - 1) / T, T, 0, stream>>>(dst, degs, E);

    // 2) norms
    vsgc_norm_kernel<<<(N + T - 1) / T, T, 0, stream>>>(degs, nl1, n05, n_05, N);

    // 3) GEMM (N assumed multiple of 16; 100000 = 6250*16)
    vsgc_gemm_wmma_kernel<<<N / 16, 128, 0, stream>>>(feat, W, bias, nl1, h, h_init);

    // 4) K=4 propagation steps; last one writes straight to d_out
    const long long totE = (long long)E * OUT_DIM;
    const int gridE = (int)((totE + T - 1) / T);
    const int gridN = (int)((NF + T - 1) / T);
    for (int k = 0; k < 4; ++k) {
        vsgc_zero_kernel<<<gridN, T, 0, stream>>>(agg, NF);
        vsgc_scatter_kernel<<<gridE, T, 0, stream>>>(src, dst, h, n_05, agg, totE);
        float* outp = (k == 3) ? (float*)d_out : h;
        vsgc_update_kernel<<<gridN, T, 0, stream>>>(agg, h, h_init, nl1, n05, outp, NF);
    }
}